// PointNet2Classify_34763465294635
// MI455X (gfx1250) — compile-verified
//
#include <hip/hip_runtime.h>
#include <hip/hip_bf16.h>
#include <stdint.h>

typedef __attribute__((ext_vector_type(2))) float v2f;
typedef __attribute__((ext_vector_type(8))) float v8f;

// ---------------- WMMA helper: D = A(16x4) * B(4x16) + C, fp32 ----------------
__device__ __forceinline__ v8f wmma_f32_4(v2f a, v2f b, v8f c) {
  return __builtin_amdgcn_wmma_f32_16x16x4_f32(false, a, false, b, (short)0, c, false, false);
}

// Order-preserving float <-> uint encoding for atomic min/max on floats.
__device__ __forceinline__ unsigned f2ord(float f) {
  unsigned u = __float_as_uint(f);
  return (u >> 31) ? ~u : (u | 0x80000000u);
}
__device__ __forceinline__ float ord2f(unsigned u) {
  return (u & 0x80000000u) ? __uint_as_float(u & 0x7FFFFFFFu) : __uint_as_float(~u);
}

// Relative-coordinate tail channel (c in 0..3, channel 3 is zero padding).
__device__ __forceinline__ float relc(const float* Pb, const float* cen, int ns, int cc) {
  return (cc < 3) ? (Pb[(size_t)ns * 3 + cc] - cen[cc]) : 0.f;
}

// ---------------- Farthest point sampling: one block per batch ----------------
__global__ void fps_kernel(const float* __restrict__ pos, int N, int m,
                           int* __restrict__ idx_out, float* __restrict__ cen_out) {
  const int T = 256;
  int b = blockIdx.x, t = threadIdx.x;
  const float* P = pos + (size_t)b * N * 3;
  int PT = N / T;                      // 16 (N=4096) or 8 (N=2048)
  float px[16], py[16], pz[16], md[16];
  for (int i = 0; i < PT; ++i) {
    int p = t + i * T;
    px[i] = P[p * 3 + 0]; py[i] = P[p * 3 + 1]; pz[i] = P[p * 3 + 2];
    md[i] = 3.4e38f;
  }
  __shared__ float rb[T];
  __shared__ int   ri[T];
  __shared__ float sp[3];
  for (int it = 0; it < m; ++it) {
    float best = -1.f; int bi = 0x7FFFFFFF;
    for (int i = 0; i < PT; ++i) {
      if (md[i] > best) { best = md[i]; bi = t + i * T; }
    }
    rb[t] = best; ri[t] = bi;
    __syncthreads();
    for (int s = T / 2; s > 0; s >>= 1) {
      if (t < s) {
        float ob = rb[t + s]; int oi = ri[t + s];
        if (ob > rb[t] || (ob == rb[t] && oi < ri[t])) { rb[t] = ob; ri[t] = oi; }
      }
      __syncthreads();
    }
    int sel = ri[0];
    if (t == (sel % T)) {
      int i = sel / T;
      sp[0] = px[i]; sp[1] = py[i]; sp[2] = pz[i];
    }
    __syncthreads();
    float sx = sp[0], sy = sp[1], sz = sp[2];
    if (t == 0) {
      idx_out[(size_t)b * m + it] = sel;
      cen_out[((size_t)b * m + it) * 3 + 0] = sx;
      cen_out[((size_t)b * m + it) * 3 + 1] = sy;
      cen_out[((size_t)b * m + it) * 3 + 2] = sz;
    }
    for (int i = 0; i < PT; ++i) {
      float dx = px[i] - sx, dy = py[i] - sy, dz = pz[i] - sz;
      md[i] = fminf(md[i], dx * dx + dy * dy + dz * dz);
    }
    __syncthreads();
  }
}

// ---------------- Radius neighbors (up to 64), one wave per center ----------------
__global__ void nbr_kernel(const float* __restrict__ pos, const float* __restrict__ centers,
                           int Npts, int m, float r2,
                           int* __restrict__ nbrs, unsigned* __restrict__ validcnt) {
  int gw   = (blockIdx.x * blockDim.x + threadIdx.x) >> 5;
  int lane = threadIdx.x & 31;
  int b = gw / m, j = gw % m;
  const float* C = centers + ((size_t)b * m + j) * 3;
  float cx = C[0], cy = C[1], cz = C[2];
  const float* P = pos + (size_t)b * Npts * 3;
  int* out = nbrs + ((size_t)b * m + j) * 64;
  int cnt = 0;
  for (int base = 0; base < Npts && cnt < 64; base += 32) {
    int p = base + lane;
    float dx = P[p * 3 + 0] - cx, dy = P[p * 3 + 1] - cy, dz = P[p * 3 + 2] - cz;
    float d2 = dx * dx + dy * dy + dz * dz;
    bool in = (d2 <= r2);
    unsigned mask = (unsigned)__ballot(in);
    int before = __popc(mask & ((1u << lane) - 1u));
    int ppos = cnt + before;
    if (in && ppos < 64) out[ppos] = p;
    cnt += __popc(mask);
  }
  if (cnt > 64) cnt = 64;
  for (int q = cnt + lane; q < 64; q += 32) out[q] = -1;
  if (lane == 0) atomicAdd(validcnt, (unsigned)cnt);
}

// ---------------- SA phase 1: GEMM1 (gathered) + masked BN stats of relu output ----------------
template <int C1>
__global__ void sa_phase1(const float* __restrict__ xin, int CX,
                          const float* __restrict__ pos, const float* __restrict__ centers,
                          const int* __restrict__ nbrs, int m, int Npts, int CIN,
                          const float* __restrict__ w1, const float* __restrict__ b1,
                          float* __restrict__ gsum, float* __restrict__ gsq) {
  __shared__ int   s_nbr[4][16];
  __shared__ float s_cen[4][3];
  __shared__ float s_sum[C1], s_sq[C1];
  int t = threadIdx.x, lane = t & 31, w = t >> 5;
  for (int c = t; c < C1; c += 128) { s_sum[c] = 0.f; s_sq[c] = 0.f; }
  size_t rowbase = ((size_t)blockIdx.x * 4 + w) * 16;
  int center = (int)(rowbase >> 6);           // 64 neighbor rows per center
  int b = center / m;
  int krow = (int)(rowbase & 63);
  if (lane < 16) s_nbr[w][lane] = nbrs[(size_t)center * 64 + krow + lane];
  if (lane < 3)  s_cen[w][lane] = centers[(size_t)center * 3 + lane];
  __syncthreads();
  const float* Pb = pos + (size_t)b * Npts * 3;
  const float* Xb = xin + (size_t)b * Npts * CX;   // unused when CX==0
  float cen[3] = { s_cen[w][0], s_cen[w][1], s_cen[w][2] };
  int r = lane & 15;
  int hi2 = (lane >> 4) * 2;                  // 0 for lanes 0-15, 2 for lanes 16-31
  int nbr_r = s_nbr[w][r];
  int ns = (nbr_r < 0) ? 0 : nbr_r;           // invalid rows feed garbage, masked later
  for (int nt = 0; nt < C1 / 16; ++nt) {
    int n0 = nt * 16;
    v8f acc = {0.f, 0.f, 0.f, 0.f, 0.f, 0.f, 0.f, 0.f};
    // main: gathered x channels, no guards
#pragma unroll 4
    for (int kt = 0; kt < CX; kt += 4) {
      int kk = kt + hi2;
      v2f av, bv;
      av.x = Xb[(size_t)ns * CX + kk];
      av.y = Xb[(size_t)ns * CX + kk + 1];
      bv.x = w1[(size_t)kk       * C1 + n0 + r];
      bv.y = w1[(size_t)(kk + 1) * C1 + n0 + r];
      acc = wmma_f32_4(av, bv, acc);
    }
    // tail: relative coordinates (3 channels) + zero pad
    {
      int kk = CX + hi2;
      v2f av, bv;
      av.x = relc(Pb, cen, ns, hi2);
      av.y = relc(Pb, cen, ns, hi2 + 1);
      bv.x = (kk     < CIN) ? w1[(size_t)kk       * C1 + n0 + r] : 0.f;
      bv.y = (kk + 1 < CIN) ? w1[(size_t)(kk + 1) * C1 + n0 + r] : 0.f;
      acc = wmma_f32_4(av, bv, acc);
    }
    float bias = b1[n0 + r];
    float psum = 0.f, psq = 0.f;
    for (int i = 0; i < 8; ++i) {
      float v = fmaxf(acc[i] + bias, 0.f);
      if (s_nbr[w][i + hi2 * 4] >= 0) { psum += v; psq += v * v; }
    }
    atomicAdd(&s_sum[n0 + r], psum);
    atomicAdd(&s_sq[n0 + r], psq);
  }
  __syncthreads();
  for (int c = t; c < C1; c += 128) {
    atomicAdd(&gsum[c], s_sum[c]);
    atomicAdd(&gsq[c], s_sq[c]);
  }
}

// ---------------- SA phase 2: GEMM1 -> BN1 -> GEMM2 -> stats2 + masked min/max aggregation ----------------
template <int C1, int C2>
__global__ void sa_phase2(const float* __restrict__ xin, int CX,
                          const float* __restrict__ pos, const float* __restrict__ centers,
                          const int* __restrict__ nbrs, int m, int Npts, int CIN,
                          const float* __restrict__ w1, const float* __restrict__ b1,
                          const float* __restrict__ mu1, const float* __restrict__ rs1,
                          const float* __restrict__ g1, const float* __restrict__ be1,
                          const float* __restrict__ w2, const float* __restrict__ b2,
                          float* __restrict__ gsum, float* __restrict__ gsq,
                          unsigned* __restrict__ mmax, unsigned* __restrict__ mmin) {
  __shared__ int   s_nbr[4][16];
  __shared__ float s_cen[4][3];
  __shared__ float s_sum[C2], s_sq[C2];
  __shared__ float s_a1[4][16][C1];
  int t = threadIdx.x, lane = t & 31, w = t >> 5;
  for (int c = t; c < C2; c += 128) { s_sum[c] = 0.f; s_sq[c] = 0.f; }
  size_t rowbase = ((size_t)blockIdx.x * 4 + w) * 16;
  int center = (int)(rowbase >> 6);
  int b = center / m;
  int krow = (int)(rowbase & 63);
  if (lane < 16) s_nbr[w][lane] = nbrs[(size_t)center * 64 + krow + lane];
  if (lane < 3)  s_cen[w][lane] = centers[(size_t)center * 3 + lane];
  __syncthreads();
  const float* Pb = pos + (size_t)b * Npts * 3;
  const float* Xb = xin + (size_t)b * Npts * CX;   // unused when CX==0
  float cen[3] = { s_cen[w][0], s_cen[w][1], s_cen[w][2] };
  int r = lane & 15;
  int hi2 = (lane >> 4) * 2;
  int nbr_r = s_nbr[w][r];
  int ns = (nbr_r < 0) ? 0 : nbr_r;
  // GEMM1 + BN1 affine -> LDS tile (layer-2 input)
  for (int nt = 0; nt < C1 / 16; ++nt) {
    int n0 = nt * 16;
    v8f acc = {0.f, 0.f, 0.f, 0.f, 0.f, 0.f, 0.f, 0.f};
#pragma unroll 4
    for (int kt = 0; kt < CX; kt += 4) {
      int kk = kt + hi2;
      v2f av, bv;
      av.x = Xb[(size_t)ns * CX + kk];
      av.y = Xb[(size_t)ns * CX + kk + 1];
      bv.x = w1[(size_t)kk       * C1 + n0 + r];
      bv.y = w1[(size_t)(kk + 1) * C1 + n0 + r];
      acc = wmma_f32_4(av, bv, acc);
    }
    {
      int kk = CX + hi2;
      v2f av, bv;
      av.x = relc(Pb, cen, ns, hi2);
      av.y = relc(Pb, cen, ns, hi2 + 1);
      bv.x = (kk     < CIN) ? w1[(size_t)kk       * C1 + n0 + r] : 0.f;
      bv.y = (kk + 1 < CIN) ? w1[(size_t)(kk + 1) * C1 + n0 + r] : 0.f;
      acc = wmma_f32_4(av, bv, acc);
    }
    int c = n0 + r;
    float bias = b1[c];
    float sl = g1[c] * rs1[c];
    float off = be1[c] - sl * mu1[c];
    for (int i = 0; i < 8; ++i)
      s_a1[w][i + hi2 * 4][c] = sl * fmaxf(acc[i] + bias, 0.f) + off;
  }
  __syncthreads();
  // GEMM2 + bias/relu + masked stats + masked raw min/max per (center, channel)
  for (int nt = 0; nt < C2 / 16; ++nt) {
    int n0 = nt * 16;
    v8f acc = {0.f, 0.f, 0.f, 0.f, 0.f, 0.f, 0.f, 0.f};
#pragma unroll 8
    for (int kt = 0; kt < C1; kt += 4) {
      int kk = kt + hi2;
      v2f av, bv;
      av.x = s_a1[w][r][kk];
      av.y = s_a1[w][r][kk + 1];
      bv.x = w2[(size_t)kk       * C2 + n0 + r];
      bv.y = w2[(size_t)(kk + 1) * C2 + n0 + r];
      acc = wmma_f32_4(av, bv, acc);
    }
    float bias = b2[n0 + r];
    float psum = 0.f, psq = 0.f, vmax = -3.4e38f, vmin = 3.4e38f;
    bool any = false;
    for (int i = 0; i < 8; ++i) {
      float v = fmaxf(acc[i] + bias, 0.f);
      if (s_nbr[w][i + hi2 * 4] >= 0) {
        psum += v; psq += v * v;
        vmax = fmaxf(vmax, v); vmin = fminf(vmin, v);
        any = true;
      }
    }
    atomicAdd(&s_sum[n0 + r], psum);
    atomicAdd(&s_sq[n0 + r], psq);
    if (any) {
      size_t oi = (size_t)center * C2 + n0 + r;
      atomicMax(&mmax[oi], f2ord(vmax));
      atomicMin(&mmin[oi], f2ord(vmin));
    }
  }
  __syncthreads();
  for (int c = t; c < C2; c += 128) {
    atomicAdd(&gsum[c], s_sum[c]);
    atomicAdd(&gsq[c], s_sq[c]);
  }
}

// ---------------- BN stat finalize: mu, rsqrt(var + eps) ----------------
__global__ void finalize_stats(const float* __restrict__ sum, const float* __restrict__ sq,
                               const unsigned* __restrict__ cntp, float cfall, int C,
                               float* __restrict__ mu, float* __restrict__ rs) {
  int c = blockIdx.x * blockDim.x + threadIdx.x;
  if (c >= C) return;
  float cnt = cntp ? (float)(*cntp) : cfall;
  float mean = sum[c] / cnt;
  float var  = sq[c] / cnt - mean * mean;
  mu[c] = mean;
  rs[c] = rsqrtf(var + 1e-5f);
}

// ---------------- Resolve: pick max/min by BN slope sign, apply affine (+optional relu) ----------------
__global__ void resolve_bn_max(const unsigned* __restrict__ mmax, const unsigned* __restrict__ mmin,
                               const float* __restrict__ mu, const float* __restrict__ rs,
                               const float* __restrict__ g, const float* __restrict__ be,
                               int C, size_t total, int do_relu, float* __restrict__ out) {
  size_t i = (size_t)blockIdx.x * blockDim.x + threadIdx.x;
  if (i >= total) return;
  int c = (int)(i % C);
  float sl = g[c] * rs[c];
  float v = ord2f((sl >= 0.f) ? mmax[i] : mmin[i]);
  float h = sl * (v - mu[c]) + be[c];
  out[i] = do_relu ? fmaxf(h, 0.f) : h;
}

// ---------------- Global MLP 131->256 with stats + per-(b,c) min/max over 512 points ----------------
__global__ void ga_kernel(const float* __restrict__ x2, const float* __restrict__ pos2,
                          const float* __restrict__ wgt, const float* __restrict__ bias,
                          float* __restrict__ gsum, float* __restrict__ gsq,
                          unsigned* __restrict__ mmax, unsigned* __restrict__ mmin) {
  const int CIN = 131, COUT = 256, NP = 512, CX = 128;
  __shared__ float s_sum[COUT], s_sq[COUT];
  int t = threadIdx.x, lane = t & 31, w = t >> 5;
  for (int c = t; c < COUT; c += 128) { s_sum[c] = 0.f; s_sq[c] = 0.f; }
  __syncthreads();
  size_t rowbase = ((size_t)blockIdx.x * 4 + w) * 16;
  int b  = (int)(rowbase / NP);
  int j0 = (int)(rowbase % NP);
  int r = lane & 15, hi2 = (lane >> 4) * 2;
  int row = j0 + r;
  const float* Xb = x2   + (size_t)b * NP * CX;
  const float* Pb = pos2 + (size_t)b * NP * 3;
  for (int nt = 0; nt < COUT / 16; ++nt) {
    int n0 = nt * 16;
    v8f acc = {0.f, 0.f, 0.f, 0.f, 0.f, 0.f, 0.f, 0.f};
#pragma unroll 4
    for (int kt = 0; kt < CX; kt += 4) {
      int kk = kt + hi2;
      v2f av, bv;
      av.x = Xb[(size_t)row * CX + kk];
      av.y = Xb[(size_t)row * CX + kk + 1];
      bv.x = wgt[(size_t)kk       * COUT + n0 + r];
      bv.y = wgt[(size_t)(kk + 1) * COUT + n0 + r];
      acc = wmma_f32_4(av, bv, acc);
    }
    {   // tail: absolute pos2 (3 channels) + zero pad
      int kk = CX + hi2;
      v2f av, bv;
      av.x = (hi2     < 3) ? Pb[(size_t)row * 3 + hi2]     : 0.f;
      av.y = (hi2 + 1 < 3) ? Pb[(size_t)row * 3 + hi2 + 1] : 0.f;
      bv.x = (kk     < CIN) ? wgt[(size_t)kk       * COUT + n0 + r] : 0.f;
      bv.y = (kk + 1 < CIN) ? wgt[(size_t)(kk + 1) * COUT + n0 + r] : 0.f;
      acc = wmma_f32_4(av, bv, acc);
    }
    float bb = bias[n0 + r];
    float psum = 0.f, psq = 0.f, vmax = -3.4e38f, vmin = 3.4e38f;
    for (int i = 0; i < 8; ++i) {
      float v = fmaxf(acc[i] + bb, 0.f);
      psum += v; psq += v * v;
      vmax = fmaxf(vmax, v); vmin = fminf(vmin, v);
    }
    atomicAdd(&s_sum[n0 + r], psum);
    atomicAdd(&s_sq[n0 + r], psq);
    size_t oi = (size_t)b * COUT + n0 + r;
    atomicMax(&mmax[oi], f2ord(vmax));
    atomicMin(&mmin[oi], f2ord(vmin));
  }
  __syncthreads();
  for (int c = t; c < COUT; c += 128) {
    atomicAdd(&gsum[c], s_sum[c]);
    atomicAdd(&gsq[c], s_sq[c]);
  }
}

// ---------------- Final FCs: relu(pooled@W0+b0) @ W1 + b1 ----------------
__global__ void fc_kernel(const float* __restrict__ pooled,
                          const float* __restrict__ w0, const float* __restrict__ b0,
                          const float* __restrict__ w1, const float* __restrict__ b1,
                          float* __restrict__ out) {
  int b = blockIdx.x, t = threadIdx.x;
  __shared__ float sp[256], sh[256];
  sp[t] = pooled[b * 256 + t];
  __syncthreads();
  float acc = b0[t];
  for (int k = 0; k < 256; ++k) acc += sp[k] * w0[k * 256 + t];
  sh[t] = fmaxf(acc, 0.f);
  __syncthreads();
  if (t < 10) {
    float o = b1[t];
    for (int k = 0; k < 256; ++k) o += sh[k] * w1[k * 10 + t];
    out[b * 10 + t] = o;
  }
}

// =====================================================================================
extern "C" void kernel_launch(void* const* d_in, const int* in_sizes, int n_in,
                              void* d_out, int out_size, void* d_ws, size_t ws_size,
                              hipStream_t stream) {
  (void)in_sizes; (void)n_in; (void)out_size; (void)ws_size;
  const float* pos   = (const float*)d_in[0];
  const float* s0w1  = (const float*)d_in[1];
  const float* s0b1  = (const float*)d_in[2];
  const float* s0g1  = (const float*)d_in[3];
  const float* s0be1 = (const float*)d_in[4];
  const float* s0w2  = (const float*)d_in[5];
  const float* s0b2  = (const float*)d_in[6];
  const float* s0g2  = (const float*)d_in[7];
  const float* s0be2 = (const float*)d_in[8];
  const float* s1w1  = (const float*)d_in[9];
  const float* s1b1  = (const float*)d_in[10];
  const float* s1g1  = (const float*)d_in[11];
  const float* s1be1 = (const float*)d_in[12];
  const float* s1w2  = (const float*)d_in[13];
  const float* s1b2  = (const float*)d_in[14];
  const float* s1g2  = (const float*)d_in[15];
  const float* s1be2 = (const float*)d_in[16];
  const float* gaw   = (const float*)d_in[17];
  const float* gab   = (const float*)d_in[18];
  const float* gag   = (const float*)d_in[19];
  const float* gabe  = (const float*)d_in[20];
  const float* l0w   = (const float*)d_in[21];
  const float* l0b   = (const float*)d_in[22];
  const float* l1w   = (const float*)d_in[23];
  const float* l1b   = (const float*)d_in[24];
  float* out = (float*)d_out;

  const int B = 8, N0 = 4096, M0 = 2048, M1 = 512, K = 64;

  // ---- carve workspace ----
  char* base = (char*)d_ws;
  size_t off = 0;
  auto carve = [&](size_t bytes) -> void* {
    void* p = base + off;
    off = (off + bytes + 255) & ~(size_t)255;
    return p;
  };
  int*      idx0   = (int*)     carve((size_t)B * M0 * 4);
  float*    cen1   = (float*)   carve((size_t)B * M0 * 3 * 4);
  int*      nbr0   = (int*)     carve((size_t)B * M0 * K * 4);
  int*      idx1   = (int*)     carve((size_t)B * M1 * 4);
  float*    cen2   = (float*)   carve((size_t)B * M1 * 3 * 4);
  int*      nbr1   = (int*)     carve((size_t)B * M1 * K * 4);
  unsigned* cnts   = (unsigned*)carve(2 * 4);
  float*    stats  = (float*)   carve((size_t)5 * 4 * 256 * 4);
  unsigned* mm0max = (unsigned*)carve((size_t)B * M0 * 128 * 4);
  unsigned* mm0min = (unsigned*)carve((size_t)B * M0 * 128 * 4);
  float*    x1     = (float*)   carve((size_t)B * M0 * 128 * 4);
  unsigned* mm1max = (unsigned*)carve((size_t)B * M1 * 128 * 4);
  unsigned* mm1min = (unsigned*)carve((size_t)B * M1 * 128 * 4);
  float*    x2     = (float*)   carve((size_t)B * M1 * 128 * 4);
  unsigned* mmgmax = (unsigned*)carve((size_t)B * 256 * 4);
  unsigned* mmgmin = (unsigned*)carve((size_t)B * 256 * 4);
  float*    pooled = (float*)   carve((size_t)B * 256 * 4);

  auto S = [&](int L, int which) { return stats + ((size_t)L * 4 + which) * 256; };

  // ---- per-call initialization of accumulators (graph-capture safe) ----
  (void)hipMemsetAsync(cnts, 0, 8, stream);
  (void)hipMemsetAsync(stats, 0, (size_t)5 * 4 * 256 * 4, stream);
  (void)hipMemsetAsync(mm0max, 0x00, (size_t)B * M0 * 128 * 4, stream);
  (void)hipMemsetAsync(mm0min, 0xFF, (size_t)B * M0 * 128 * 4, stream);
  (void)hipMemsetAsync(mm1max, 0x00, (size_t)B * M1 * 128 * 4, stream);
  (void)hipMemsetAsync(mm1min, 0xFF, (size_t)B * M1 * 128 * 4, stream);
  (void)hipMemsetAsync(mmgmax, 0x00, (size_t)B * 256 * 4, stream);
  (void)hipMemsetAsync(mmgmin, 0xFF, (size_t)B * 256 * 4, stream);

  // ---- SA0: pos[8,4096,3] -> x1[8,2048,128], cen1[8,2048,3] ----
  fps_kernel<<<B, 256, 0, stream>>>(pos, N0, M0, idx0, cen1);
  nbr_kernel<<<(B * M0) / 8, 256, 0, stream>>>(pos, cen1, N0, M0, 0.04f, nbr0, &cnts[0]);
  sa_phase1<64><<<(B * M0 * K) / 64, 128, 0, stream>>>(
      pos /*unused*/, 0, pos, cen1, nbr0, M0, N0, 3, s0w1, s0b1, S(0, 0), S(0, 1));
  finalize_stats<<<1, 64, 0, stream>>>(S(0, 0), S(0, 1), &cnts[0], 0.f, 64, S(0, 2), S(0, 3));
  sa_phase2<64, 128><<<(B * M0 * K) / 64, 128, 0, stream>>>(
      pos /*unused*/, 0, pos, cen1, nbr0, M0, N0, 3,
      s0w1, s0b1, S(0, 2), S(0, 3), s0g1, s0be1,
      s0w2, s0b2, S(1, 0), S(1, 1), mm0max, mm0min);
  finalize_stats<<<1, 128, 0, stream>>>(S(1, 0), S(1, 1), &cnts[0], 0.f, 128, S(1, 2), S(1, 3));
  resolve_bn_max<<<(B * M0 * 128) / 256, 256, 0, stream>>>(
      mm0max, mm0min, S(1, 2), S(1, 3), s0g2, s0be2, 128, (size_t)B * M0 * 128, 0, x1);

  // ---- SA1: (x1, cen1) -> x2[8,512,128], cen2[8,512,3] ----
  fps_kernel<<<B, 256, 0, stream>>>(cen1, M0, M1, idx1, cen2);
  nbr_kernel<<<(B * M1) / 8, 256, 0, stream>>>(cen1, cen2, M0, M1, 0.16f, nbr1, &cnts[1]);
  sa_phase1<128><<<(B * M1 * K) / 64, 128, 0, stream>>>(
      x1, 128, cen1, cen2, nbr1, M1, M0, 131, s1w1, s1b1, S(2, 0), S(2, 1));
  finalize_stats<<<1, 128, 0, stream>>>(S(2, 0), S(2, 1), &cnts[1], 0.f, 128, S(2, 2), S(2, 3));
  sa_phase2<128, 128><<<(B * M1 * K) / 64, 128, 0, stream>>>(
      x1, 128, cen1, cen2, nbr1, M1, M0, 131,
      s1w1, s1b1, S(2, 2), S(2, 3), s1g1, s1be1,
      s1w2, s1b2, S(3, 0), S(3, 1), mm1max, mm1min);
  finalize_stats<<<1, 128, 0, stream>>>(S(3, 0), S(3, 1), &cnts[1], 0.f, 128, S(3, 2), S(3, 3));
  resolve_bn_max<<<(B * M1 * 128) / 256, 256, 0, stream>>>(
      mm1max, mm1min, S(3, 2), S(3, 3), s1g2, s1be2, 128, (size_t)B * M1 * 128, 0, x2);

  // ---- Global MLP + max pool + FC head ----
  ga_kernel<<<(B * M1) / 64, 128, 0, stream>>>(x2, cen2, gaw, gab, S(4, 0), S(4, 1), mmgmax, mmgmin);
  finalize_stats<<<1, 256, 0, stream>>>(S(4, 0), S(4, 1), (const unsigned*)0,
                                        (float)(B * M1), 256, S(4, 2), S(4, 3));
  resolve_bn_max<<<(B * 256) / 256, 256, 0, stream>>>(
      mmgmax, mmgmin, S(4, 2), S(4, 3), gag, gabe, 256, (size_t)B * 256, 1, pooled);
  fc_kernel<<<B, 256, 0, stream>>>(pooled, l0w, l0b, l1w, l1b, out);
}